// SemimarkovModel_89008902242487
// MI455X (gfx1250) — compile-verified
//
#include <hip/hip_runtime.h>

// CDNA5 / gfx1250. wave32.
// out[b,t,k,c1,c2] = T[c1,c2] + L[k,c2] + window_sum(E)[b,t,k,c2]
//                    + (t==0)*init[c2] + (k==511-t && k>=1)*E[b,511,c1]
// Each 32x32 (c1,c2) tile is a rank-2 update of T:  T + ones*v^T + e*ones^T,
// computed with V_WMMA_F32_16X16X4_F32:  D = A(16x4) * B(4x16) + C
//   A[:,0]=1, A[:,1]=e ; B[0,:]=v, B[1,:]=1 ; C = T subtile.

typedef __attribute__((ext_vector_type(2))) float v2f;
typedef __attribute__((ext_vector_type(8))) float v8f;

#define BB 4
#define NN 512
#define KK 20
#define CC 32
#define NT (NN - 1)   // 511

__global__ __launch_bounds__(KK * 32) void hsmm_scores_wmma_kernel(
    const float* __restrict__ T,     // (32,32)
    const float* __restrict__ E,     // (4,512,32)
    const float* __restrict__ initv, // (32,)
    const float* __restrict__ L,     // (20,32)
    float* __restrict__ out)         // (4,511,20,32,32)
{
  __shared__ float sE[KK][CC];   // sE[k][c2] = sum_{j=t..t+k-1, j<512} E[b,j,c2]
  __shared__ float sEos[CC];     // E[b,511,:]
  __shared__ float sInit[CC];

  const int blk = blockIdx.x;
  const int b = blk / NT;
  const int t = blk - b * NT;
  const int tid = threadIdx.x;

  // ---- Phase 1: one wave builds the clipped window prefix sums in LDS ----
  if (tid < CC) {
    const int l = tid;
    const float* Eb = E + (size_t)b * NN * CC;
    float acc = 0.0f;
    sE[0][l] = 0.0f;
#pragma unroll
    for (int k = 1; k < KK; ++k) {
      const int j = t + k - 1;               // frame index, clip at N
      const float ev = (j < NN) ? Eb[(size_t)j * CC + l] : 0.0f;
      acc += ev;
      sE[k][l] = acc;
    }
    sEos[l]  = Eb[(size_t)(NN - 1) * CC + l];
    sInit[l] = initv[l];
  }
  __syncthreads();

  // ---- Phase 2: wave w handles duration k = w ----
  const int k    = tid >> 5;      // wave id == duration
  const int lane = tid & 31;
  const int half = lane >> 4;     // 0: lanes 0-15 (K=0,1), 1: lanes 16-31 (K=2,3)
  const int li   = lane & 15;

  // low = 1 for lanes 0-15, 0 for lanes 16-31: zeroes the K=2,3 slots branchlessly
  const float low = (half == 0) ? 1.0f : 0.0f;
  const bool  has_eos = (k >= 1) && (t == (NN - 1) - k);   // wave-uniform
  const bool  is_t0   = (t == 0);                          // block-uniform

  // A operand per m-subtile: lane holds A[M=li][K=0]=1, A[M=li][K=1]=e[M]
  v2f aOp[2];
#pragma unroll
  for (int ms = 0; ms < 2; ++ms) {
    const int m = ms * 16 + li;
    const float e = has_eos ? sEos[m] : 0.0f;
    aOp[ms].x = low;
    aOp[ms].y = low * e;
  }

  // B operand per n-subtile: lane holds B[K=0][N=li]=v[N], B[K=1][N=li]=1
  v2f bOp[2];
#pragma unroll
  for (int ns = 0; ns < 2; ++ns) {
    const int n = ns * 16 + li;
    float v = sE[k][n] + L[k * CC + n];
    if (is_t0) v += sInit[n];
    bOp[ns].x = low * v;
    bOp[ns].y = low;
  }

  float* outTile = out + ((((size_t)b * NT + t) * KK + k) * (CC * CC));

#pragma unroll
  for (int ms = 0; ms < 2; ++ms) {
#pragma unroll
    for (int ns = 0; ns < 2; ++ns) {
      // C/D layout: VGPR r -> row (msub*16 + half*8 + r), col (nsub*16 + li)
      const int rbase = ms * 16 + half * 8;
      const int col   = ns * 16 + li;
      v8f c;
#pragma unroll
      for (int r = 0; r < 8; ++r)
        c[r] = T[(rbase + r) * CC + col];          // L0/L2-cached, 4 KB total

      v8f d = __builtin_amdgcn_wmma_f32_16x16x4_f32(
          /*neg_a=*/false, aOp[ms],
          /*neg_b=*/false, bOp[ns],
          /*c_mod=*/(short)0, c,
          /*reuse_a=*/false, /*reuse_b=*/false);

#pragma unroll
      for (int r = 0; r < 8; ++r)
        __builtin_nontemporal_store(d[r], &outTile[(size_t)(rbase + r) * CC + col]);
    }
  }
}

extern "C" void kernel_launch(void* const* d_in, const int* in_sizes, int n_in,
                              void* d_out, int out_size, void* d_ws, size_t ws_size,
                              hipStream_t stream) {
  (void)in_sizes; (void)n_in; (void)d_ws; (void)ws_size; (void)out_size;
  const float* T     = (const float*)d_in[0];  // (32,32)
  const float* E     = (const float*)d_in[1];  // (4,512,32)
  const float* initv = (const float*)d_in[2];  // (32,)
  const float* L     = (const float*)d_in[3];  // (20,32)
  float* out = (float*)d_out;                  // (4,511,20,32,32)

  dim3 grid(BB * NT);          // 2044 blocks, one per (b, t)
  dim3 block(KK * 32);         // 20 waves, one per duration k
  hsmm_scores_wmma_kernel<<<grid, block, 0, stream>>>(T, E, initv, L, out);
}